// CycleLoss_85667417686358
// MI455X (gfx1250) — compile-verified
//
#include <hip/hip_runtime.h>
#include <stdint.h>

#define BATCH   262144
#define NSTEPS  10
#define TPB     128            // 4 waves of 32
#define ROWF    68             // padded LDS row stride in floats (17 float4, breaks bank alignment)
#define NCHUNK  15             // 60 floats = 15 float4 chunks per sample

typedef float v2f __attribute__((ext_vector_type(2)));
typedef float v8f __attribute__((ext_vector_type(8)));

// ---------------------------------------------------------------------------
// Wave-level sum using V_WMMA_F32_16X16X4_F32 (full fp32, exact).
// A = all-ones 16x4, B holds one partial per lane (other slot zero):
// every D[m,n] = colsum(n) = acc[n] + acc[n+16]; lane L holds colsum(L%16)
// in c[0]. Four xor-shuffles over the low 4 lane bits sum the 16 column sums,
// giving the full 32-lane total in every lane.
// Requires EXEC == all ones (call from full, convergent waves only).
// ---------------------------------------------------------------------------
__device__ __forceinline__ float wave_sum_wmma(float x) {
    v2f a; a[0] = 1.0f; a[1] = 1.0f;
    v2f b; b[0] = x;    b[1] = 0.0f;
    v8f c = {};
    c = __builtin_amdgcn_wmma_f32_16x16x4_f32(false, a, false, b,
                                              (short)0, c, false, false);
    float s = c[0];
    s += __shfl_xor(s, 1, 32);
    s += __shfl_xor(s, 2, 32);
    s += __shfl_xor(s, 4, 32);
    s += __shfl_xor(s, 8, 32);
    return s;
}

// euler(x,y,z) -> row-major 3x3
__device__ __forceinline__ void e2m(float ex, float ey, float ez, float R[9]) {
    float sx, cx, sy, cy, sz, cz;
    __sincosf(ex, &sx, &cx);
    __sincosf(ey, &sy, &cy);
    __sincosf(ez, &sz, &cz);
    R[0] = cz * cy; R[1] = cz * sy * sx - sz * cx; R[2] = cz * sy * cx + sz * sx;
    R[3] = sz * cy; R[4] = sz * sy * sx + cz * cx; R[5] = sz * sy * cx - cz * sx;
    R[6] = -sy;     R[7] = cy * sx;                R[8] = cy * cx;
}

// row-major 3x3 -> euler; branch-free operand selection (one atan2 per angle)
__device__ __forceinline__ void m2e(const float M[9], float& x, float& y, float& z) {
    float sy = sqrtf(M[0] * M[0] + M[3] * M[3]);
    bool sing = sy < 1e-6f;
    float xa = sing ? -M[5] : M[7];
    float xb = sing ?  M[4] : M[8];
    x = atan2f(xa, xb);
    y = atan2f(-M[6], sy);
    z = sing ? 0.0f : atan2f(M[3], M[0]);
}

// ---------------------------------------------------------------------------
// Pass A: one sample per thread; inputs staged global->LDS via CDNA5 async
// copies (coalesced: consecutive threads copy consecutive 16B chunks).
// ---------------------------------------------------------------------------
__global__ void __launch_bounds__(TPB)
cycle_loss_main(const float* __restrict__ pred, const float* __restrict__ gt,
                float* __restrict__ partial)
{
    __shared__ __align__(16) float smP[TPB * ROWF];
    __shared__ __align__(16) float smG[TPB * ROWF];
    __shared__ float wsum[TPB / 32];

    const int tid = threadIdx.x;
    const long long blockBase = (long long)blockIdx.x * TPB * 60;

    // ---- async stage: 15 iterations x 128 threads x 16B per array ----
    const uint64_t gp = (uint64_t)(uintptr_t)(const void*)(pred + blockBase);
    const uint64_t gg = (uint64_t)(uintptr_t)(const void*)(gt + blockBase);
    // low 32 bits of a flat shared pointer == workgroup-relative LDS byte addr
    const uint32_t lp = (uint32_t)(uintptr_t)(const void*)&smP[0];
    const uint32_t lg = (uint32_t)(uintptr_t)(const void*)&smG[0];
#pragma unroll
    for (int it = 0; it < NCHUNK; ++it) {
        int c = it * TPB + tid;       // global chunk id (contiguous in memory)
        int s = c / NCHUNK;           // sample within block
        int p = c - s * NCHUNK;       // float4 within sample
        uint32_t loff = (uint32_t)(s * (ROWF * 4) + p * 16);
        uint32_t ldP = lp + loff;
        uint64_t gaP = gp + (uint64_t)c * 16u;
        asm volatile("global_load_async_to_lds_b128 %0, %1, off"
                     :: "v"(ldP), "v"(gaP) : "memory");
        uint32_t ldG = lg + loff;
        uint64_t gaG = gg + (uint64_t)c * 16u;
        asm volatile("global_load_async_to_lds_b128 %0, %1, off"
                     :: "v"(ldG), "v"(gaG) : "memory");
    }
    asm volatile("s_wait_asynccnt 0x0" ::: "memory");
    __syncthreads();

    const int rb = tid * ROWF;
    float acc = 0.0f;

    // ---------------- translation recurrence ----------------
    float vp[3], vg[3], Sp[3], Sg[3];
#pragma unroll
    for (int k = 0; k < 3; ++k) {
        vp[k] = smP[rb + k];
        vg[k] = smG[rb + k];
        float d = vp[k] - vg[k];
        acc += d * d;
        Sp[k] = 0.0f; Sg[k] = 0.0f;
    }
#pragma unroll
    for (int i = 1; i < NSTEPS; ++i) {
#pragma unroll
        for (int k = 0; k < 3; ++k) {
            if (i >= 2) {                       // compile-time condition
                Sp[k] += smP[rb + (i - 1) * 6 + k];
                Sg[k] += smG[rb + (i - 1) * 6 + k];
            }
            vp[k] = 2.0f * vp[k] + Sp[k];
            vg[k] = 2.0f * vg[k] + Sg[k];
            float d = vp[k] - vg[k];
            acc += d * d;
        }
    }

    // ---------------- rotation part (elementwise cumprod) ----------------
    float R1p[9], R1g[9], Pp[9], Pg[9];
    e2m(smP[rb + 9], smP[rb + 10], smP[rb + 11], R1p);   // step-1 angles
    e2m(smG[rb + 9], smG[rb + 10], smG[rb + 11], R1g);
#pragma unroll
    for (int k = 0; k < 9; ++k) { Pp[k] = 1.0f; Pg[k] = 1.0f; }

#pragma unroll
    for (int s = 0; s < NSTEPS; ++s) {
        float Mp[9], Mg[9];
#pragma unroll
        for (int k = 0; k < 9; ++k) { Mp[k] = R1p[k] * Pp[k]; Mg[k] = R1g[k] * Pg[k]; }
        float xp, yp, zp, xg, yg, zg;
        m2e(Mp, xp, yp, zp);
        m2e(Mg, xg, yg, zg);
        float d;
        d = xp - xg; acc += d * d;
        d = yp - yg; acc += d * d;
        d = zp - zg; acc += d * d;

        if (s < NSTEPS - 1) {                   // update running cumprod with R[s]
            float Rp[9], Rg[9];
            e2m(smP[rb + s * 6 + 3], smP[rb + s * 6 + 4], smP[rb + s * 6 + 5], Rp);
            e2m(smG[rb + s * 6 + 3], smG[rb + s * 6 + 4], smG[rb + s * 6 + 5], Rg);
#pragma unroll
            for (int k = 0; k < 9; ++k) { Pp[k] *= Rp[k]; Pg[k] *= Rg[k]; }
        }
    }

    // ---------------- block reduction (WMMA + xor shuffles + LDS) ----------------
    float w = wave_sum_wmma(acc);
    int lane = tid & 31, wv = tid >> 5;
    if (lane == 0) wsum[wv] = w;
    __syncthreads();
    if (tid == 0) {
        float b = 0.0f;
#pragma unroll
        for (int i = 0; i < TPB / 32; ++i) b += wsum[i];
        partial[blockIdx.x] = b;
    }
}

// ---------------------------------------------------------------------------
// Pass B: reduce block partials, apply final scale, write scalar loss.
// ---------------------------------------------------------------------------
__global__ void __launch_bounds__(TPB)
cycle_loss_final(const float* __restrict__ partial, float* __restrict__ out,
                 int n, float scale)
{
    __shared__ float wsum[TPB / 32];
    float acc = 0.0f;
    for (int i = threadIdx.x; i < n; i += TPB) acc += partial[i];
    float w = wave_sum_wmma(acc);       // all threads reach here -> EXEC full
    int lane = threadIdx.x & 31, wv = threadIdx.x >> 5;
    if (lane == 0) wsum[wv] = w;
    __syncthreads();
    if (threadIdx.x == 0) {
        float b = 0.0f;
#pragma unroll
        for (int i = 0; i < TPB / 32; ++i) b += wsum[i];
        out[0] = b * scale;
    }
}

extern "C" void kernel_launch(void* const* d_in, const int* in_sizes, int n_in,
                              void* d_out, int out_size, void* d_ws, size_t ws_size,
                              hipStream_t stream) {
    const float* pred = (const float*)d_in[0];
    const float* gt   = (const float*)d_in[1];
    float* partial    = (float*)d_ws;           // 2048 floats = 8 KB scratch
    float* out        = (float*)d_out;

    const int nblocks = BATCH / TPB;            // 2048, exact
    hipLaunchKernelGGL(cycle_loss_main, dim3(nblocks), dim3(TPB), 0, stream,
                       pred, gt, partial);

    // loss = mean(sq, over BATCH*10*2*3 elems) / BATCH
    const double sc = 1.0 / ((double)BATCH * 60.0 * (double)BATCH);
    hipLaunchKernelGGL(cycle_loss_final, dim3(1), dim3(TPB), 0, stream,
                       partial, out, nblocks, (float)sc);
}